// GHMC_1580547966503
// MI455X (gfx1250) — compile-verified
//
#include <hip/hip_runtime.h>
#include <cstdint>

#define TPB 256
#define BINS 10
#define ITERS_MAX 10   // 4*ITERS_MAX + 1 = 41 <= 63, fits 6-bit packed per-bin counts

// ---------------------------------------------------------------------------
// CDNA5 async global->LDS copy (tracked by ASYNCcnt). vdst = 32-bit LDS byte
// address VGPR, vaddr = 64-bit global address VGPR pair. Stream is read-once,
// so use the non-temporal hint.
// ---------------------------------------------------------------------------
__device__ __forceinline__ uint32_t lds_byte_addr(const void* p) {
    // Flat LDS aperture: addr[31:0] is the workgroup-relative LDS byte offset.
    return (uint32_t)(uintptr_t)p;
}

__device__ __forceinline__ void async_copy16(uint32_t lds_off, const void* gptr) {
    asm volatile("global_load_async_to_lds_b128 %0, %1, off th:TH_LOAD_NT"
                 :
                 : "v"(lds_off), "v"((uint64_t)(uintptr_t)gptr)
                 : "memory");
}

#if __has_builtin(__builtin_amdgcn_s_wait_asynccnt)
#define ASYNC_WAIT(n) __builtin_amdgcn_s_wait_asynccnt(n)
#else
#define ASYNC_WAIT(n) asm volatile("s_wait_asynccnt %0" :: "i"(n) : "memory")
#endif

// ---------------------------------------------------------------------------
// Per-element GHM-C work: bce = softplus(x) - x*t ; g = |sigmoid(x) - t| ;
// bin = clamp(floor(10 g), 0, 9). One v_exp serves sigmoid and softplus; the
// bin-sum scatter goes to this thread's PRIVATE LDS bin array via ds_add_f32
// (DS pipe, deterministic: each lane's atomics hit only its own slots).
// Valid-count is kept as a packed u64 register (10 x 6-bit fields).
// ---------------------------------------------------------------------------
__device__ __forceinline__ void ghmc_elem(float x, float t, float w,
                                          float* __restrict__ myBins,
                                          unsigned long long& cnt) {
    const bool valid = w > 0.0f;
    const float ax  = fabsf(x);
    const float ea  = __expf(-ax);                         // exp(-|x|), v_exp_f32
    const float lg  = __logf(1.0f + ea);                   // ~log1p(ea), v_log_f32
    const float bce = fmaxf(x, 0.0f) - x * t + lg;         // softplus(x) - x*t
    const float r   = __builtin_amdgcn_rcpf(1.0f + ea);    // v_rcp_f32
    const float sg  = (x >= 0.0f) ? r : ea * r;            // sigmoid(x)
    const float g   = fabsf(sg - t);
    int bin = (int)(g * (float)BINS);                      // g >= 0, trunc == floor
    bin = bin > (BINS - 1) ? (BINS - 1) : bin;

    cnt += valid ? (1ULL << (6 * bin)) : 0ULL;
    const float bv = valid ? bce : 0.0f;                   // add 0 when invalid
    __hip_atomic_fetch_add(&myBins[bin], bv, __ATOMIC_RELAXED,
                           __HIP_MEMORY_SCOPE_WORKGROUP);
}

// ---------------------------------------------------------------------------
// Kernel 1: streaming pass. Double-buffered async-to-LDS staging of all three
// streams; per-thread LDS bin sums + packed register counts; deterministic
// block tree reduction. Output per block: 10 f32 sums + 10 u32 counts.
// ---------------------------------------------------------------------------
__global__ __launch_bounds__(TPB) void ghmc_partial(
    const float* __restrict__ pred, const float* __restrict__ target,
    const float* __restrict__ lw, float* __restrict__ ws,
    int nVec4, int nElem, int iters)
{
    __shared__ float4 bP[2][TPB];
    __shared__ float4 bT[2][TPB];
    __shared__ float4 bW[2][TPB];
    __shared__ float    sB[TPB * BINS];   // per-thread private bin sums
    __shared__ unsigned sU[TPB * BINS];   // per-thread bin counts (unpacked late)

    const int tid = threadIdx.x;
    float* myB = &sB[tid * BINS];
#pragma unroll
    for (int b = 0; b < BINS; ++b) myB[b] = 0.0f;  // own slots; LDS is in-order
    unsigned long long cnt = 0ULL;

    const long long base = (long long)blockIdx.x * TPB * iters;
    const float4* p4 = (const float4*)pred;
    const float4* t4 = (const float4*)target;
    const float4* w4 = (const float4*)lw;

    auto issue = [&](int it, int buf) {
        long long idx = base + (long long)it * TPB + tid;
        long long cidx = (idx < nVec4) ? idx : (nVec4 > 0 ? (long long)nVec4 - 1 : 0);
        async_copy16(lds_byte_addr(&bP[buf][tid]), p4 + cidx);
        async_copy16(lds_byte_addr(&bT[buf][tid]), t4 + cidx);
        async_copy16(lds_byte_addr(&bW[buf][tid]), w4 + cidx);
    };

    if (iters > 0 && nVec4 > 0) {
        issue(0, 0);
        for (int it = 0; it < iters; ++it) {
            if (it + 1 < iters) {
                issue(it + 1, (it + 1) & 1);
                ASYNC_WAIT(3);   // iter 'it' loads done; next iter's 3 in flight
            } else {
                ASYNC_WAIT(0);
            }
            const long long idx = base + (long long)it * TPB + tid;
            const float4 p = bP[it & 1][tid];
            const float4 t = bT[it & 1][tid];
            const float4 w = bW[it & 1][tid];
            if (idx < nVec4) {
                ghmc_elem(p.x, t.x, w.x, myB, cnt);
                ghmc_elem(p.y, t.y, w.y, myB, cnt);
                ghmc_elem(p.z, t.z, w.z, myB, cnt);
                ghmc_elem(p.w, t.w, w.w, myB, cnt);
            }
        }
    }

    // Tail (nElem % 4), handled by the first threads of block 0.
    if (blockIdx.x == 0) {
        const int tail = nElem - nVec4 * 4;
        if (tid < tail) {
            const int i = nVec4 * 4 + tid;
            ghmc_elem(pred[i], target[i], lw[i], myB, cnt);
        }
    }

    // Unpack packed counts, then deterministic block tree reduction.
#pragma unroll
    for (int b = 0; b < BINS; ++b)
        sU[tid * BINS + b] = (unsigned)((cnt >> (6 * b)) & 63ULL);
    __syncthreads();
    for (int s = TPB / 2; s > 0; s >>= 1) {
        if (tid < s) {
#pragma unroll
            for (int b = 0; b < BINS; ++b) {
                sB[tid * BINS + b] += sB[(tid + s) * BINS + b];
                sU[tid * BINS + b] += sU[(tid + s) * BINS + b];
            }
        }
        __syncthreads();
    }
    if (tid == 0) {
        float* of = ws + (size_t)blockIdx.x * (2 * BINS);
        unsigned* ou = (unsigned*)(of + BINS);
#pragma unroll
        for (int b = 0; b < BINS; ++b) { of[b] = sB[b]; ou[b] = sU[b]; }
    }
}

// ---------------------------------------------------------------------------
// Kernel 2: deterministic final reduction over G block-partials.
// loss = (1/n) * sum_b S_b / counts_b   over non-empty bins (tot cancels).
// ---------------------------------------------------------------------------
__global__ __launch_bounds__(TPB) void ghmc_final(
    const float* __restrict__ ws, int G, float* __restrict__ out)
{
    __shared__ float    sF[TPB * BINS];
    __shared__ unsigned sU[TPB * BINS];
    const int tid = threadIdx.x;

    float acc[BINS];
    unsigned c[BINS];
#pragma unroll
    for (int b = 0; b < BINS; ++b) { acc[b] = 0.0f; c[b] = 0u; }

    for (int r = tid; r < G; r += TPB) {
        const float* of = ws + (size_t)r * (2 * BINS);
        const unsigned* ou = (const unsigned*)(of + BINS);
#pragma unroll
        for (int b = 0; b < BINS; ++b) { acc[b] += of[b]; c[b] += ou[b]; }
    }

#pragma unroll
    for (int b = 0; b < BINS; ++b) {
        sF[tid * BINS + b] = acc[b];
        sU[tid * BINS + b] = c[b];
    }
    __syncthreads();
    for (int s = TPB / 2; s > 0; s >>= 1) {
        if (tid < s) {
#pragma unroll
            for (int b = 0; b < BINS; ++b) {
                sF[tid * BINS + b] += sF[(tid + s) * BINS + b];
                sU[tid * BINS + b] += sU[(tid + s) * BINS + b];
            }
        }
        __syncthreads();
    }
    if (tid == 0) {
        float loss = 0.0f;
        int n = 0;
        for (int b = 0; b < BINS; ++b) {
            if (sU[b] > 0u) { ++n; loss += sF[b] / (float)sU[b]; }
        }
        out[0] = (n > 0) ? (loss / (float)n) : 0.0f;   // LOSS_WEIGHT = 1
    }
}

// ---------------------------------------------------------------------------
extern "C" void kernel_launch(void* const* d_in, const int* in_sizes, int n_in,
                              void* d_out, int out_size, void* d_ws, size_t ws_size,
                              hipStream_t stream) {
    const float* pred   = (const float*)d_in[0];
    const float* target = (const float*)d_in[1];
    const float* lw     = (const float*)d_in[2];
    float* out = (float*)d_out;
    float* ws  = (float*)d_ws;

    const int nElem = in_sizes[0];          // 262144 * 80 = 20,971,520
    const int nVec4 = nElem / 4;            // 5,242,880
    const int iters = ITERS_MAX;            // 10 float4 chunks per thread
    const int chunksPerBlock = TPB * iters; // 2560
    int G = (nVec4 + chunksPerBlock - 1) / chunksPerBlock;  // 2048 here
    if (G < 1) G = 1;
    // workspace: G * 20 * 4 bytes = 160 KB for the default problem size

    ghmc_partial<<<G, TPB, 0, stream>>>(pred, target, lw, ws, nVec4, nElem, iters);
    ghmc_final<<<1, TPB, 0, stream>>>(ws, G, out);
}